// ChebConvTimeFFT_13288628814253
// MI455X (gfx1250) — compile-verified
//
#include <hip/hip_runtime.h>
#include <hip/hip_bf16.h>
#include <math.h>

typedef __attribute__((ext_vector_type(2))) float v2f;
typedef __attribute__((ext_vector_type(8))) float v8f;

#define HZ 32           // horizon
#define FF 16           // in channels
#define GG 16           // out channels
#define KK 3            // Chebyshev order
#define RR 17           // onesided rfft bins
#define RS 34           // RR * 2 (re/im interleaved slices)
static constexpr float INV_SQRT_H = 0.17677669529663687f; // 1/sqrt(32)
static constexpr float TWO_PI = 6.283185307179586f;

// ---------------- utility kernels ----------------
__global__ void zero_f32(float* p, long long n) {
  long long i = (long long)blockIdx.x * blockDim.x + threadIdx.x;
  long long stride = (long long)gridDim.x * blockDim.x;
  for (; i < n; i += stride) p[i] = 0.0f;
}

__global__ void negcopy_f32(const float* __restrict__ s, float* __restrict__ d, long long n) {
  long long i = (long long)blockIdx.x * blockDim.x + threadIdx.x;
  long long stride = (long long)gridDim.x * blockDim.x;
  for (; i < n; i += stride) d[i] = -s[i];
}

// ---------------- graph Laplacian ----------------
__global__ void degree_kernel(const long long* __restrict__ ei, long long E, float* deg) {
  long long e = (long long)blockIdx.x * blockDim.x + threadIdx.x;
  if (e >= E) return;
  long long r = ei[e], c = ei[E + e];
  if (r != c) atomicAdd(&deg[r], 1.0f);
}

__global__ void lap_kernel(const long long* __restrict__ ei, const float* __restrict__ ew,
                           const float* __restrict__ deg, long long E, float* lap) {
  long long e = (long long)blockIdx.x * blockDim.x + threadIdx.x;
  if (e >= E) return;
  long long r = ei[e], c = ei[E + e];
  float dr = deg[r], dc = deg[c];
  float ir = dr > 0.0f ? rsqrtf(dr) : 0.0f;
  float ic = dc > 0.0f ? rsqrtf(dc) : 0.0f;
  lap[e] = (r != c) ? (-ir * ew[e] * ic) : 0.0f;
}

// ---------------- forward rFFT (ortho), 16 nodes per block ----------------
// x: [N, H, F] -> tx0 slices: [rs][N][F], rs = 2*r + s
__global__ void rfft_kernel(const float* __restrict__ x, float* __restrict__ tx0, int N) {
  __shared__ float xs[16 * HZ * FF]; // 8192 floats = 32 KB
  __shared__ float ct[32], st[32];
  int tid = threadIdx.x; // 256 threads
  if (tid < 32) {
    float a = TWO_PI * (float)tid / 32.0f;
    ct[tid] = cosf(a);
    st[tid] = sinf(a);
  }
  int n0 = blockIdx.x * 16;
  const float* src = x + (long long)n0 * (HZ * FF);
  int limit = (N - n0 >= 16 ? 16 : (N - n0)) * HZ * FF;
  for (int i = tid; i < 16 * HZ * FF; i += 256) xs[i] = (i < limit) ? src[i] : 0.0f;
  __syncthreads();
  long long NF = (long long)N * FF;
  for (int o = tid; o < 16 * RR * FF; o += 256) {
    int nl  = o / (RR * FF);
    int rem = o - nl * (RR * FF);
    int r = rem >> 4;
    int f = rem & 15;
    int n = n0 + nl;
    if (n >= N) continue;
    float re = 0.0f, im = 0.0f;
    const float* px = &xs[nl * (HZ * FF) + f];
    #pragma unroll
    for (int t = 0; t < HZ; ++t) {
      float v = px[t * FF];
      int a = (r * t) & 31;
      re = fmaf(v, ct[a], re);
      im = fmaf(-v, st[a], im);
    }
    tx0[(long long)(2 * r)     * NF + (long long)n * FF + f] = re * INV_SQRT_H;
    tx0[(long long)(2 * r + 1) * NF + (long long)n * FF + f] = im * INV_SQRT_H;
  }
}

// ---------------- Laplacian scatter: dst[rs][col][:] += scale*lap[e]*src[rs][row][:] ----------------
// One thread per edge; loops all 34 slices internally so edge data is read once.
// Atomics are non-returning (STOREcnt-tracked) and pipeline deeply.
__global__ void scatter_kernel(const long long* __restrict__ ei, const float* __restrict__ lap,
                               const float* __restrict__ src, float* dst,
                               long long E, long long NF, float scale) {
  long long e = (long long)blockIdx.x * blockDim.x + threadIdx.x;
  if (e >= E) return;
  float l = lap[e] * scale;
  if (l == 0.0f) return;
  long long r = ei[e], c = ei[E + e];
  const float* s = src + r * FF;
  float*       d = dst + c * FF;
  for (int rs = 0; rs < RS; ++rs) {
    const float* sp = s + (long long)rs * NF;
    float*       dp = d + (long long)rs * NF;
    #pragma unroll
    for (int f = 0; f < FF; f += 4) {
      float4 v = *(const float4*)(sp + f);
      atomicAdd(dp + f + 0, l * v.x);
      atomicAdd(dp + f + 1, l * v.y);
      atomicAdd(dp + f + 2, l * v.z);
      atomicAdd(dp + f + 3, l * v.w);
    }
  }
}

// ---------------- WMMA channel mix: yh[rs][n][g] = sum_k sum_f Txk[rs][n][f] * W[k][f][g][rs] ----------------
// grid = (WMMA_BLK, RS); 128 threads = 4 waves. Each wave grid-strides over 16-node
// tiles, keeping the 12 B (weight) fragments resident in registers across tiles.
// 12 chained V_WMMA_F32_16X16X4_F32 per tile (3 orders x 4 K-chunks).
#define WMMA_BLK 64
__global__ void wmma_mix_kernel(const float* __restrict__ tx0,
                                const float* __restrict__ tx1,
                                const float* __restrict__ tx2,
                                const float* __restrict__ w,
                                float* __restrict__ yh, int N) {
  int lane = threadIdx.x & 31;
  int wave = threadIdx.x >> 5;
  int rs   = blockIdx.y;
  int ntiles = N >> 4;
  long long NF = (long long)N * FF;
  int g  = lane & 15;
  int hi = lane >> 4;                    // 0 or 1 (upper half-wave holds K+2 rows)
  int ml = lane & 15;                    // A-matrix row within tile

  // B fragments: lane holds W[k][f0..f0+1][g][rs], f0 = 4*kc + 2*hi (ISA 16x4 f32 layout)
  v2f bw[KK][4];
  #pragma unroll
  for (int k = 0; k < KK; ++k) {
    #pragma unroll
    for (int kc = 0; kc < 4; ++kc) {
      int f0 = 4 * kc + 2 * hi;
      bw[k][kc].x = w[(((long long)(k * FF + f0    ) * GG + g) * RS) + rs];
      bw[k][kc].y = w[(((long long)(k * FF + f0 + 1) * GG + g) * RS) + rs];
    }
  }

  const float* txs[KK] = { tx0, tx1, tx2 };
  float* out = yh + (long long)rs * NF;
  int wstride = (int)gridDim.x * 4;

  for (int tile = (int)blockIdx.x * 4 + wave; tile < ntiles; tile += wstride) {
    int m0 = tile << 4;

    // prefetch next tile's A rows (lowers to global_prefetch_b8)
    int tnext = tile + wstride;
    if (tnext < ntiles) {
      #pragma unroll
      for (int k = 0; k < KK; ++k)
        __builtin_prefetch(txs[k] + (long long)rs * NF +
                           (long long)((tnext << 4) + ml) * FF, 0, 0);
    }

    v8f c = {};
    #pragma unroll
    for (int k = 0; k < KK; ++k) {
      const float* base = txs[k] + (long long)rs * NF + (long long)(m0 + ml) * FF + 2 * hi;
      #pragma unroll
      for (int kc = 0; kc < 4; ++kc) {
        v2f a = *(const v2f*)(base + 4 * kc);  // K = 4*kc + 2*hi + {0,1}
        c = __builtin_amdgcn_wmma_f32_16x16x4_f32(
            /*neg_a=*/false, a, /*neg_b=*/false, bw[k][kc],
            /*c_mod=*/(short)0, c, /*reuse_a=*/false, /*reuse_b=*/false);
      }
    }

    // D layout: VGPR j -> (M = j + 8*hi, N = lane%16)
    #pragma unroll
    for (int j = 0; j < 8; ++j) {
      int node = m0 + j + 8 * hi;
      out[(long long)node * FF + g] = c[j];
    }
  }
}

// ---------------- inverse rFFT (ortho) + bias: out[n][t][g] ----------------
__global__ void irfft_kernel(const float* __restrict__ yh, const float* __restrict__ bias,
                             float* __restrict__ out, int N) {
  __shared__ float ys[RS * 16 * GG]; // 8704 floats = 34.8 KB : ys[rs][nl][g]
  __shared__ float ct[32], st[32], bsh[GG];
  int tid = threadIdx.x; // 256 threads
  if (tid < 32) {
    float a = TWO_PI * (float)tid / 32.0f;
    ct[tid] = cosf(a);
    st[tid] = sinf(a);
  }
  if (tid < GG) bsh[tid] = bias[tid];
  int n0 = blockIdx.x * 16;
  long long NF = (long long)N * GG;
  int nvalid = (N - n0 >= 16) ? 16 : (N - n0);
  for (int rs = 0; rs < RS; ++rs) {
    if (tid < nvalid * GG)
      ys[rs * 256 + tid] = yh[(long long)rs * NF + (long long)n0 * GG + tid];
  }
  __syncthreads();
  for (int o = tid; o < 16 * HZ * GG; o += 256) {
    int nl = o >> 9;          // /512
    int t  = (o >> 4) & 31;
    int g  = o & 15;
    if (n0 + nl >= N) continue;
    const float* yb = &ys[nl * GG + g];  // rs stride = 256
    float acc = yb[0];                                   // r=0 (real)
    acc += ((t & 1) ? -1.0f : 1.0f) * yb[32 * 256];      // r=16 Nyquist (real)
    #pragma unroll
    for (int r = 1; r < 16; ++r) {
      int a = (r * t) & 31;
      acc += 2.0f * (yb[(2 * r) * 256] * ct[a] - yb[(2 * r + 1) * 256] * st[a]);
    }
    out[(long long)(n0 + nl) * (HZ * GG) + t * GG + g] = acc * INV_SQRT_H + bsh[g];
  }
}

// ---------------- launch ----------------
extern "C" void kernel_launch(void* const* d_in, const int* in_sizes, int n_in,
                              void* d_out, int out_size, void* d_ws, size_t ws_size,
                              hipStream_t stream) {
  const float*     x    = (const float*)d_in[0];      // [N,H,F]
  const long long* ei   = (const long long*)d_in[1];  // [2,E] int64
  const float*     ew   = (const float*)d_in[2];      // [E]
  const float*     w    = (const float*)d_in[3];      // [K,F,G,R,2]
  const float*     bias = (const float*)d_in[4];      // [G]
  float*           out  = (float*)d_out;              // [N,H,G]

  int N = in_sizes[0] / (HZ * FF);
  long long E = (long long)in_sizes[2];
  long long NF = (long long)N * FF;
  long long S  = (long long)RS * NF;                  // floats per Tx buffer

  float* tx0 = (float*)d_ws;
  float* tx1 = tx0 + S;
  float* tx2 = tx1 + S;
  float* yh  = tx2 + S;
  float* deg = yh + S;
  float* lap = deg + N;

  int nblk16 = (N + 15) / 16;
  long long eblk = (E + 255) / 256;

  zero_f32<<<2048, 256, 0, stream>>>(deg, (long long)N);
  zero_f32<<<4096, 256, 0, stream>>>(tx1, S);
  degree_kernel<<<(unsigned)eblk, 256, 0, stream>>>(ei, E, deg);
  lap_kernel<<<(unsigned)eblk, 256, 0, stream>>>(ei, ew, deg, E, lap);

  rfft_kernel<<<nblk16, 256, 0, stream>>>(x, tx0, N);

  scatter_kernel<<<(unsigned)eblk, 256, 0, stream>>>(ei, lap, tx0, tx1, E, NF, 1.0f); // Tx1 = L Tx0
  negcopy_f32<<<4096, 256, 0, stream>>>(tx0, tx2, S);                                  // Tx2 = -Tx0
  scatter_kernel<<<(unsigned)eblk, 256, 0, stream>>>(ei, lap, tx1, tx2, E, NF, 2.0f); // Tx2 += 2 L Tx1

  dim3 wgrid(WMMA_BLK, RS);
  wmma_mix_kernel<<<wgrid, 128, 0, stream>>>(tx0, tx1, tx2, w, yh, N);

  irfft_kernel<<<nblk16, 256, 0, stream>>>(yh, bias, out, N);
}